// YoloLossModel_58935541236092
// MI455X (gfx1250) — compile-verified
//
#include <hip/hip_runtime.h>
#include <hip/hip_bf16.h>
#include <stdint.h>

// ---------------------------------------------------------------------------
// YOLO loss, CDNA5 (gfx1250). Bandwidth-bound: stream P,T (193 MB) once via
// async global->LDS copies, compute per-cell loss in registers, deterministic
// two-level reduction. No matmul structure -> WMMA not applicable; the
// gfx1250-specific win is GLOBAL_LOAD_ASYNC_TO_LDS_B128 + s_wait_asynccnt.
// ---------------------------------------------------------------------------

#define NCH 30
#define CELLS_PER_BLOCK 256
#define THREADS 256
#define TILE_FLOATS (CELLS_PER_BLOCK * NCH)       // 7680 floats per tensor tile
#define NV4 (TILE_FLOATS / 4)                     // 1920 float4 transfers

#if defined(__has_builtin)
#  if __has_builtin(__builtin_amdgcn_global_load_async_to_lds_b128)
#    define HAVE_ASYNC_B128 1
#  endif
#  if __has_builtin(__builtin_amdgcn_s_wait_asynccnt)
#    define HAVE_WAIT_ASYNC 1
#  endif
#endif

// Builtin signature (from hipcc diagnostic): param0 = v4i in addrspace(1),
// param1 = v4i in addrspace(3), then imm offset, imm cpol.
typedef int v4i __attribute__((vector_size(16)));
typedef __attribute__((address_space(1))) v4i gv4i;
typedef __attribute__((address_space(3))) v4i lv4i;

__device__ __forceinline__ void async_copy16(const float* __restrict__ gsrc,
                                             float* __restrict__ ldst) {
#if defined(HAVE_ASYNC_B128)
    __builtin_amdgcn_global_load_async_to_lds_b128(
        (gv4i*)(float*)gsrc,   // drop const in generic AS, then -> addrspace(1)
        (lv4i*)ldst,           // generic -> addrspace(3) (true addrspacecast)
        0, 0);
#else
    // Inline-asm form per CDNA5 ISA (GV mode, SADDR=off).
    unsigned loff = (unsigned)(size_t)((__attribute__((address_space(3))) float*)ldst);
    unsigned long long ga = (unsigned long long)(uintptr_t)gsrc;
    asm volatile("global_load_async_to_lds_b128 %0, %1, off"
                 :: "v"(loff), "v"(ga) : "memory");
#endif
}

__device__ __forceinline__ void wait_async_all() {
#if defined(HAVE_WAIT_ASYNC)
    __builtin_amdgcn_s_wait_asynccnt(0);
#else
    asm volatile("s_wait_asynccnt 0x0" ::: "memory");
#endif
}

__global__ __launch_bounds__(THREADS)
void yolo_loss_partial(const float* __restrict__ P,
                       const float* __restrict__ T,
                       float* __restrict__ partial) {
    __shared__ __align__(16) float lp[TILE_FLOATS];
    __shared__ __align__(16) float lt[TILE_FLOATS];

    const int tid = threadIdx.x;
    const size_t fbase = (size_t)blockIdx.x * CELLS_PER_BLOCK * NCH;

    // Stage 256 cells of P and T into LDS: 1920 x b128 per tensor,
    // perfectly coalesced, no VGPR round-trip (ASYNCcnt-tracked).
#pragma unroll
    for (int r = 0; r < (NV4 + THREADS - 1) / THREADS; ++r) {
        int i = r * THREADS + tid;
        if (i < NV4) {
            async_copy16(P + fbase + (size_t)i * 4, &lp[i * 4]);
            async_copy16(T + fbase + (size_t)i * 4, &lt[i * 4]);
        }
    }
    wait_async_all();     // each wave's async writes complete...
    __syncthreads();      // ...before any wave reads LDS

    // ---- per-cell loss (lane tid owns cell tid; stride-30 reads are
    //      bank-conflict-free under wave32: gcd(30,64)=2 -> 32 distinct banks)
    const float* p = &lp[tid * NCH];
    const float* t = &lt[tid * NCH];

    const float inv_s = 1.0f / 7.0f;

    const float t4 = t[4];
    const float tcx = t[0] * inv_s, tcy = t[1] * inv_s;
    const float tw = t[2], th = t[3];
    const float tx1 = tcx - 0.5f * tw, ty1 = tcy - 0.5f * th;
    const float tx2 = tcx + 0.5f * tw, ty2 = tcy + 0.5f * th;
    const float areaT = (tx2 - tx1) * (ty2 - ty1);

    float iou0 = 0.0f, iou1 = 0.0f;
#pragma unroll
    for (int j = 0; j < 2; ++j) {
        const float* b = p + 5 * j;
        float cx = b[0] * inv_s, cy = b[1] * inv_s;
        float w = b[2], h = b[3];
        float x1 = cx - 0.5f * w, y1 = cy - 0.5f * h;
        float x2 = cx + 0.5f * w, y2 = cy + 0.5f * h;
        float lx = fmaxf(x1, tx1), ly = fmaxf(y1, ty1);
        float rx = fminf(x2, tx2), ry = fminf(y2, ty2);
        float iw = fmaxf(rx - lx, 0.0f), ih = fmaxf(ry - ly, 0.0f);
        float inter = iw * ih;
        float areaP = (x2 - x1) * (y2 - y1);
        float iou = inter / (areaP + areaT - inter);
        if (j == 0) iou0 = iou; else iou1 = iou;
    }
    const int   idx  = (iou1 > iou0) ? 1 : 0;   // argmax tie -> box 0 (matches jnp.argmax)
    const float ioub = fmaxf(iou0, iou1);
    const int   sb   = 5 * idx;

    const float dx = p[sb + 0] - t[sb + 0];
    const float dy = p[sb + 1] - t[sb + 1];
    const float lXY = dx * dx + dy * dy;

    const float sw = sqrtf(p[sb + 2]) - sqrtf(t[sb + 2]);   // wh in [0.05,0.95] > 0
    const float sh = sqrtf(p[sb + 3]) - sqrtf(t[sb + 3]);
    const float lWH = sw * sw + sh * sh;

    const float dconf = p[sb + 4] - ioub;
    const float lObj = dconf * dconf;

    const float d4 = p[4] - t4;
    const float d9 = p[9] - t[9];
    const float lNO = d4 * d4 + d9 * d9;

    float lCls = 0.0f;
#pragma unroll
    for (int c = 10; c < 30; ++c) {
        float d = p[c] - t[c];
        lCls += d * d;
    }

    const float objf   = (t4 == 1.0f) ? 1.0f : 0.0f;  // T conf channel is exact 0/1
    const float noobjf = (t4 == 0.0f) ? 1.0f : 0.0f;
    float L = objf * (5.0f * (lXY + lWH) + lObj + lCls) + 0.5f * noobjf * lNO;

    // ---- deterministic in-block tree reduction (reuse lp after barrier)
    __syncthreads();
    lp[tid] = L;
    __syncthreads();
#pragma unroll
    for (int s2 = THREADS / 2; s2 > 0; s2 >>= 1) {
        if (tid < s2) lp[tid] += lp[tid + s2];
        __syncthreads();
    }
    if (tid == 0) partial[blockIdx.x] = lp[0];
}

__global__ __launch_bounds__(THREADS)
void yolo_loss_final(const float* __restrict__ partial,
                     float* __restrict__ out, int nb, float inv_batch) {
    __shared__ float sm[THREADS];
    float s = 0.0f;
    for (int i = threadIdx.x; i < nb; i += THREADS) s += partial[i];  // fixed order
    sm[threadIdx.x] = s;
    __syncthreads();
#pragma unroll
    for (int k = THREADS / 2; k > 0; k >>= 1) {
        if (threadIdx.x < k) sm[threadIdx.x] += sm[threadIdx.x + k];
        __syncthreads();
    }
    if (threadIdx.x == 0) out[0] = sm[0] * inv_batch;
}

extern "C" void kernel_launch(void* const* d_in, const int* in_sizes, int n_in,
                              void* d_out, int out_size, void* d_ws, size_t ws_size,
                              hipStream_t stream) {
    const float* P = (const float*)d_in[0];
    const float* T = (const float*)d_in[1];
    float* out = (float*)d_out;
    float* partial = (float*)d_ws;

    const int cells = in_sizes[0] / NCH;              // 802816
    const int nb = cells / CELLS_PER_BLOCK;           // 3136 (exact)
    const int batch = in_sizes[0] / (7 * 7 * NCH);    // 16384

    yolo_loss_partial<<<nb, THREADS, 0, stream>>>(P, T, partial);
    yolo_loss_final<<<1, THREADS, 0, stream>>>(partial, out, nb, 1.0f / (float)batch);
}